// DecoderRNN_8598524526807
// MI455X (gfx1250) — compile-verified
//
#include <hip/hip_runtime.h>
#include <hip/hip_bf16.h>
#include <math.h>

#define HIDDEN 2048
#define VOCAB  50257

typedef __attribute__((ext_vector_type(2))) float v2f;
typedef __attribute__((ext_vector_type(8))) float v8f;

// ---------------------------------------------------------------------------
// k0: x = relu(embedding[idx])
// ---------------------------------------------------------------------------
__global__ void embed_relu_kernel(const int* __restrict__ idx,
                                  const float* __restrict__ emb,
                                  float* __restrict__ x) {
  int i = blockIdx.x * blockDim.x + threadIdx.x;
  if (i < HIDDEN) {
    float v = emb[(size_t)idx[0] * HIDDEN + i];
    x[i] = v > 0.0f ? v : 0.0f;
  }
}

// ---------------------------------------------------------------------------
// WMMA-based GEMV: y[r] = dot(W[r, 0:2048], v) + bias[r],  W row-major R x 2048
//
// One wave computes 16 consecutive rows. A-fragment (16x4 f32): lane holds
// W[base + lane%16, k + 2*(lane/16) + {0,1}] -> a single global_load_b64 per
// iteration, non-temporal (weights are streamed once, no reuse).
// B-fragment (4x16 f32) is built so every column equals the vector chunk
// (B[k,n] = v[k0+k] for all n): per the ISA C/D-style striping this is the
// same per-lane K rule as A, i.e. a broadcast ds_load_b64 of v[k + 2*half].
// Then every column of D = the matvec; lanes 0 and 16 hold rows base+0..7 /
// base+8..15 in their 8 accumulator VGPRs.
// ---------------------------------------------------------------------------
__global__ __launch_bounds__(256) void gemv_wmma_kernel(
    const float* __restrict__ W, const float* __restrict__ v,
    const float* __restrict__ bias, float* __restrict__ y, int R) {
  __shared__ float sv[HIDDEN];
  for (int i = threadIdx.x; i < HIDDEN; i += blockDim.x) sv[i] = v[i];
  __syncthreads();

  const int lane  = threadIdx.x & 31;
  const int wave  = threadIdx.x >> 5;
  const int group = blockIdx.x * (blockDim.x >> 5) + wave;  // 16-row group
  const int base  = group * 16;
  if (base >= R) return;  // wave-uniform: EXEC stays all-ones for WMMA

  const int m    = lane & 15;
  const int half = lane >> 4;           // 0: K={0,1}, 1: K={2,3}
  int row = base + m;
  row = (row < R) ? row : (R - 1);      // clamp tail loads (stores are guarded)

  const float* wrow = W + (size_t)row * HIDDEN + 2 * half;
  const float* svp  = sv + 2 * half;

  v8f acc = {};
#pragma unroll 8
  for (int k = 0; k < HIDDEN; k += 4) {
    v2f a = __builtin_nontemporal_load((const v2f*)(wrow + k));
    v2f b = *(const v2f*)(svp + k);     // LDS broadcast (2 groups: X, X+8)
    acc = __builtin_amdgcn_wmma_f32_16x16x4_f32(
        /*neg_a=*/false, a, /*neg_b=*/false, b,
        /*c_mod=*/(short)0, acc, /*reuse_a=*/false, /*reuse_b=*/false);
  }

  // D layout: VGPR r holds M=r (lanes 0-15) / M=8+r (lanes 16-31).
  // Columns are identical, so lane 0 -> rows base+0..7, lane 16 -> base+8..15.
  if (m == 0) {
    int r0 = base + 8 * half;
#pragma unroll
    for (int i = 0; i < 8; ++i) {
      int r = r0 + i;
      if (r < R) y[r] = acc[i] + bias[r];
    }
  }
}

// ---------------------------------------------------------------------------
// k2: GRU gate math  (gi/gh already include biases)
// ---------------------------------------------------------------------------
__device__ __forceinline__ float sigmoidf_(float x) {
  return 1.0f / (1.0f + expf(-x));
}

__global__ void gru_gates_kernel(const float* __restrict__ gi,
                                 const float* __restrict__ gh,
                                 const float* __restrict__ h,
                                 float* __restrict__ hn) {
  int j = blockIdx.x * blockDim.x + threadIdx.x;
  if (j < HIDDEN) {
    float r = sigmoidf_(gi[j]            + gh[j]);
    float z = sigmoidf_(gi[j + HIDDEN]   + gh[j + HIDDEN]);
    float n = tanhf    (gi[j + 2*HIDDEN] + r * gh[j + 2*HIDDEN]);
    hn[j] = (1.0f - z) * n + z * h[j];
  }
}

// ---------------------------------------------------------------------------
// k4: single-block logsumexp reduction over the logits (L2-resident, 201 KB)
// red[0] = max, red[1] = log(sum(exp(l - max)))
// ---------------------------------------------------------------------------
__global__ __launch_bounds__(1024) void lse_reduce_kernel(
    const float* __restrict__ logits, float* __restrict__ red, int n) {
  __shared__ float s[1024];
  const int tid = threadIdx.x;

  float m = -3.402823466e+38f;
  for (int i = tid; i < n; i += 1024) m = fmaxf(m, logits[i]);
  s[tid] = m;
  __syncthreads();
  for (int off = 512; off > 0; off >>= 1) {
    if (tid < off) s[tid] = fmaxf(s[tid], s[tid + off]);
    __syncthreads();
  }
  m = s[0];
  __syncthreads();

  float sum = 0.0f;
  for (int i = tid; i < n; i += 1024) sum += expf(logits[i] - m);
  s[tid] = sum;
  __syncthreads();
  for (int off = 512; off > 0; off >>= 1) {
    if (tid < off) s[tid] += s[tid + off];
    __syncthreads();
  }
  if (tid == 0) {
    red[0] = m;
    red[1] = logf(s[0]);
  }
}

// k5: logp[i] = logits[i] - max - log(sum)   (in place on d_out)
__global__ void lsm_norm_kernel(float* __restrict__ logits,
                                const float* __restrict__ red, int n) {
  int i = blockIdx.x * blockDim.x + threadIdx.x;
  if (i < n) logits[i] = logits[i] - red[0] - red[1];
}

// ---------------------------------------------------------------------------
// Launch: k0 -> gemv(w_ih,x) -> gemv(w_hh,h) -> gates -> gemv(out_w,h_new)
//         -> lse reduce -> normalize.  All on `stream`, graph-capture safe.
// ---------------------------------------------------------------------------
extern "C" void kernel_launch(void* const* d_in, const int* in_sizes, int n_in,
                              void* d_out, int out_size, void* d_ws, size_t ws_size,
                              hipStream_t stream) {
  const int*   idx    = (const int*)  d_in[0];
  const float* hidden = (const float*)d_in[1];
  const float* emb    = (const float*)d_in[2];
  const float* w_ih   = (const float*)d_in[3];
  const float* w_hh   = (const float*)d_in[4];
  const float* b_ih   = (const float*)d_in[5];
  const float* b_hh   = (const float*)d_in[6];
  const float* out_w  = (const float*)d_in[7];
  const float* out_b  = (const float*)d_in[8];

  float* out    = (float*)d_out;         // [0,VOCAB): logp ; [VOCAB,+H): h_new
  float* ws     = (float*)d_ws;
  float* x      = ws;                    // 2048
  float* gi     = ws + HIDDEN;           // 6144
  float* gh     = ws + HIDDEN + 3*HIDDEN;// 6144
  float* red    = ws + HIDDEN + 6*HIDDEN;// 2
  float* logits = out;
  float* h_new  = out + VOCAB;

  // k0: embedding gather + relu
  embed_relu_kernel<<<(HIDDEN + 255) / 256, 256, 0, stream>>>(idx, emb, x);

  // k1a/k1b: gate pre-activations (6144 rows each -> 384 groups -> 48 blocks)
  {
    int groups = (3 * HIDDEN) / 16;
    int blocks = (groups + 7) / 8;
    gemv_wmma_kernel<<<blocks, 256, 0, stream>>>(w_ih, x,      b_ih, gi, 3 * HIDDEN);
    gemv_wmma_kernel<<<blocks, 256, 0, stream>>>(w_hh, hidden, b_hh, gh, 3 * HIDDEN);
  }

  // k2: GRU gates -> h_new (second output, written directly into d_out)
  gru_gates_kernel<<<(HIDDEN + 255) / 256, 256, 0, stream>>>(gi, gh, hidden, h_new);

  // k3: output projection, 50257 rows -> 3142 groups -> 393 blocks
  {
    int groups = (VOCAB + 15) / 16;
    int blocks = (groups + 7) / 8;
    gemv_wmma_kernel<<<blocks, 256, 0, stream>>>(out_w, h_new, out_b, logits, VOCAB);
  }

  // k4/k5: log_softmax
  lse_reduce_kernel<<<1, 1024, 0, stream>>>(logits, red, VOCAB);
  lsm_norm_kernel<<<(VOCAB + 255) / 256, 256, 0, stream>>>(logits, red, VOCAB);
}